// GAT_55018531062635
// MI455X (gfx1250) — compile-verified
//
#include <hip/hip_runtime.h>
#include <math.h>

#define N_NODES   100000
#define N_EDGES   1600000
#define E_TOT     (N_EDGES + N_NODES)
#define IN_CH     128
#define HID       64
#define HEADS     2
#define FDIM      (HEADS * HID)   // 128
#define NEG_SLOPE 0.2f
#define EPS_F     1e-16f

typedef float v2f __attribute__((ext_vector_type(2)));
typedef float v8f __attribute__((ext_vector_type(8)));

// ---------------------------------------------------------------------------
// WMMA fp32 GEMM: C[M,NC] = A[M,K] @ B[K,NC]  (row-major, M % 16 == 0, NC=128)
// One wave per 16x16 C tile; 8 waves per block cover the 128-wide N dim.
// K and NC are compile-time so all operand loads use immediate IOFFSETs
// (no per-iteration 64-bit address arithmetic) and the K-loop fully unrolls
// into load-clauses + v_wmma_f32_16x16x4_f32.
// Lane layouts per ISA 7.12.2:
//   A 16x4 : lanes 0-15 rows, v0 = K(2*half), v1 = K(2*half+1)
//   B 4x16 : lanes 0-15 cols, v0 = K(2*half), v1 = K(2*half+1)
//   C 16x16: VGPR r -> row (r + 8*half), col = lane&15
// ---------------------------------------------------------------------------
template <int K, int NC>
__global__ __launch_bounds__(256) void gemm_f32_wmma(const float* __restrict__ A,
                                                     const float* __restrict__ B,
                                                     float* __restrict__ C) {
  const int lane = threadIdx.x & 31;
  const int wave = threadIdx.x >> 5;
  const int m0   = blockIdx.x * 16;
  const int n0   = wave * 16;
  const int row  = lane & 15;   // A-row / B-col within tile
  const int half = lane >> 4;   // selects K pair {0,1} vs {2,3}

  const float* Ap = A + (size_t)(m0 + row) * K + 2 * half;
  const float* Bp = B + (size_t)(2 * half) * NC + n0 + row;

  v8f acc = {};
#pragma unroll
  for (int k0 = 0; k0 < K; k0 += 4) {
    v2f a, b;
    a.x = Ap[k0 + 0];                 // imm offset k0*4
    a.y = Ap[k0 + 1];                 // imm offset k0*4+4
    b.x = Bp[(size_t)(k0 + 0) * NC];  // imm offset k0*NC*4
    b.y = Bp[(size_t)(k0 + 1) * NC];  // imm offset (k0+1)*NC*4
    acc = __builtin_amdgcn_wmma_f32_16x16x4_f32(false, a, false, b,
                                                (short)0, acc, false, false);
  }
#pragma unroll
  for (int r = 0; r < 8; ++r) {
    C[(size_t)(m0 + r + 8 * half) * NC + (n0 + row)] = acc[r];
  }
}

// ---------------------------------------------------------------------------
// Per-node attention coefficients: a_s[n,h] = <H[n,h,:], att_s[h,:]>
// ---------------------------------------------------------------------------
__global__ void att_coef_kernel(const float* __restrict__ H,
                                const float* __restrict__ att_s,
                                const float* __restrict__ att_d,
                                float* __restrict__ a_s,
                                float* __restrict__ a_d) {
  int n = blockIdx.x * blockDim.x + threadIdx.x;
  if (n >= N_NODES) return;
  const float* hp = H + (size_t)n * FDIM;
#pragma unroll
  for (int h = 0; h < HEADS; ++h) {
    float s = 0.f, d = 0.f;
    for (int c = 0; c < HID; ++c) {
      float v = hp[h * HID + c];
      s = fmaf(v, att_s[h * HID + c], s);
      d = fmaf(v, att_d[h * HID + c], d);
    }
    a_s[n * HEADS + h] = s;
    a_d[n * HEADS + h] = d;
  }
}

// Monotone float<->uint encoding so atomicMax(u32) == float max.
__device__ __forceinline__ unsigned enc_f32(float f) {
  unsigned b = __float_as_uint(f);
  return (b & 0x80000000u) ? ~b : (b | 0x80000000u);
}
__device__ __forceinline__ float dec_f32(unsigned u) {
  unsigned b = (u & 0x80000000u) ? (u & 0x7fffffffu) : ~u;
  return __uint_as_float(b);
}
__device__ __forceinline__ float leaky(float x) {
  return x > 0.f ? x : NEG_SLOPE * x;
}
__device__ __forceinline__ void edge_sd(const int* __restrict__ ei, int e,
                                        int& s, int& d) {
  if (e < N_EDGES) {
    s = ei[e];
    d = ei[N_EDGES + e];
  } else {
    s = d = e - N_EDGES;   // self loop
  }
}

// Pass 1: segment max of leaky_relu logits per (dst, head)
__global__ void edge_max_kernel(const int* __restrict__ ei,
                                const float* __restrict__ a_s,
                                const float* __restrict__ a_d,
                                unsigned* __restrict__ amax) {
  int e = blockIdx.x * blockDim.x + threadIdx.x;
  if (e >= E_TOT) return;
  int s, d;
  edge_sd(ei, e, s, d);
#pragma unroll
  for (int h = 0; h < HEADS; ++h) {
    float l = leaky(a_s[s * HEADS + h] + a_d[d * HEADS + h]);
    atomicMax(&amax[d * HEADS + h], enc_f32(l));
  }
}

// Pass 2: ex = exp(logit - max); store ex, atomic-add denominator
__global__ void edge_exp_kernel(const int* __restrict__ ei,
                                const float* __restrict__ a_s,
                                const float* __restrict__ a_d,
                                const unsigned* __restrict__ amax,
                                float* __restrict__ exb,
                                float* __restrict__ denom) {
  int e = blockIdx.x * blockDim.x + threadIdx.x;
  if (e >= E_TOT) return;
  int s, d;
  edge_sd(ei, e, s, d);
#pragma unroll
  for (int h = 0; h < HEADS; ++h) {
    float l  = leaky(a_s[s * HEADS + h] + a_d[d * HEADS + h]);
    float ex = expf(l - dec_f32(amax[d * HEADS + h]));
    exb[(size_t)e * HEADS + h] = ex;
    atomicAdd(&denom[d * HEADS + h], ex);
  }
}

// Pass 3: acc[dst] += (ex/denom) * H[src]; one wave per edge, 4 ch per lane
__global__ __launch_bounds__(256) void edge_scatter_kernel(
    const int* __restrict__ ei, const float* __restrict__ H,
    const float* __restrict__ exb, const float* __restrict__ denom,
    float* __restrict__ acc) {
  long long tid = (long long)blockIdx.x * blockDim.x + threadIdx.x;
  int e = (int)(tid >> 5);
  if (e >= E_TOT) return;
  int lane = (int)(tid & 31);
  int s, d;
  edge_sd(ei, e, s, d);
  int c0 = lane * 4;        // channels [c0, c0+3], all within one head
  int h  = c0 >> 6;
  float w = exb[(size_t)e * HEADS + h] / (denom[d * HEADS + h] + EPS_F);
  const float4 hv = *(const float4*)(H + (size_t)s * FDIM + c0);
  float* ap = acc + (size_t)d * FDIM + c0;
  atomicAdd(ap + 0, w * hv.x);
  atomicAdd(ap + 1, w * hv.y);
  atomicAdd(ap + 2, w * hv.z);
  atomicAdd(ap + 3, w * hv.w);
}

// Layer-1 epilogue: head mean + bias + ELU -> x2[N, 64]
__global__ void finalize1_kernel(const float* __restrict__ acc,
                                 const float* __restrict__ bias,
                                 float* __restrict__ xo) {
  int idx = blockIdx.x * blockDim.x + threadIdx.x;
  if (idx >= N_NODES * HID) return;
  int n = idx >> 6, c = idx & 63;
  float v = 0.5f * (acc[(size_t)n * FDIM + c] + acc[(size_t)n * FDIM + HID + c]) + bias[c];
  v = v > 0.f ? v : (expf(v) - 1.f);
  xo[idx] = v;
}

// Layer-2 epilogue fused with final linear: out[n] = elu(mean+b2) @ lin_w + lin_b
__global__ void finalize2_kernel(const float* __restrict__ acc,
                                 const float* __restrict__ bias,
                                 const float* __restrict__ lw,
                                 const float* __restrict__ lb,
                                 float* __restrict__ out) {
  int n = blockIdx.x * blockDim.x + threadIdx.x;
  if (n >= N_NODES) return;
  float o = lb[0];
  for (int c = 0; c < HID; ++c) {
    float v = 0.5f * (acc[(size_t)n * FDIM + c] + acc[(size_t)n * FDIM + HID + c]) + bias[c];
    v = v > 0.f ? v : (expf(v) - 1.f);
    o = fmaf(v, lw[c], o);
  }
  out[n] = o;
}

extern "C" void kernel_launch(void* const* d_in, const int* in_sizes, int n_in,
                              void* d_out, int out_size, void* d_ws, size_t ws_size,
                              hipStream_t stream) {
  const float* x     = (const float*)d_in[0];
  const int*   ei    = (const int*)  d_in[1];
  const float* W1    = (const float*)d_in[2];
  const float* as1   = (const float*)d_in[3];
  const float* ad1   = (const float*)d_in[4];
  const float* b1    = (const float*)d_in[5];
  const float* W2    = (const float*)d_in[6];
  const float* as2   = (const float*)d_in[7];
  const float* ad2   = (const float*)d_in[8];
  const float* b2    = (const float*)d_in[9];
  const float* lin_w = (const float*)d_in[10];
  const float* lin_b = (const float*)d_in[11];
  float* out = (float*)d_out;

  // Workspace carve-out (256B aligned), ~145 MB total.
  char* ws = (char*)d_ws;
  size_t off = 0;
  auto carve = [&](size_t bytes) -> char* {
    char* p = ws + off;
    off = (off + bytes + 255) & ~(size_t)255;
    return p;
  };
  float*    H     = (float*)   carve((size_t)N_NODES * FDIM * 4);   // node features (both layers)
  float*    acc   = (float*)   carve((size_t)N_NODES * FDIM * 4);   // scatter accumulator
  float*    x2    = (float*)   carve((size_t)N_NODES * HID * 4);    // layer-1 output
  float*    a_s   = (float*)   carve((size_t)N_NODES * HEADS * 4);
  float*    a_d   = (float*)   carve((size_t)N_NODES * HEADS * 4);
  unsigned* amax  = (unsigned*)carve((size_t)N_NODES * HEADS * 4);
  float*    denom = (float*)   carve((size_t)N_NODES * HEADS * 4);
  float*    exb   = (float*)   carve((size_t)E_TOT * HEADS * 4);

  const dim3 blk(256);
  const int g_node   = (N_NODES + 255) / 256;
  const int g_edge   = (E_TOT + 255) / 256;
  const int g_scat   = (int)(((long long)E_TOT * 32 + 255) / 256);
  const int g_gemm   = N_NODES / 16;   // 6250, exact
  const int g_fin1   = (N_NODES * HID + 255) / 256;
  const size_t nh_b  = (size_t)N_NODES * HEADS * 4;
  const size_t acc_b = (size_t)N_NODES * FDIM * 4;

  // -------------------- Layer 1 --------------------
  gemm_f32_wmma<IN_CH, FDIM><<<g_gemm, blk, 0, stream>>>(x, W1, H);
  att_coef_kernel<<<g_node, blk, 0, stream>>>(H, as1, ad1, a_s, a_d);
  hipMemsetAsync(amax, 0, nh_b, stream);
  hipMemsetAsync(denom, 0, nh_b, stream);
  hipMemsetAsync(acc, 0, acc_b, stream);
  edge_max_kernel<<<g_edge, blk, 0, stream>>>(ei, a_s, a_d, amax);
  edge_exp_kernel<<<g_edge, blk, 0, stream>>>(ei, a_s, a_d, amax, exb, denom);
  edge_scatter_kernel<<<g_scat, blk, 0, stream>>>(ei, H, exb, denom, acc);
  finalize1_kernel<<<g_fin1, blk, 0, stream>>>(acc, b1, x2);

  // -------------------- Layer 2 --------------------
  gemm_f32_wmma<HID, FDIM><<<g_gemm, blk, 0, stream>>>(x2, W2, H);
  att_coef_kernel<<<g_node, blk, 0, stream>>>(H, as2, ad2, a_s, a_d);
  hipMemsetAsync(amax, 0, nh_b, stream);
  hipMemsetAsync(denom, 0, nh_b, stream);
  hipMemsetAsync(acc, 0, acc_b, stream);
  edge_max_kernel<<<g_edge, blk, 0, stream>>>(ei, a_s, a_d, amax);
  edge_exp_kernel<<<g_edge, blk, 0, stream>>>(ei, a_s, a_d, amax, exb, denom);
  edge_scatter_kernel<<<g_scat, blk, 0, stream>>>(ei, H, exb, denom, acc);
  finalize2_kernel<<<g_node, blk, 0, stream>>>(acc, b2, lin_w, lin_b, out);
}